// FasterRCNN_61804579389974
// MI455X (gfx1250) — compile-verified
//
#include <hip/hip_runtime.h>

// ---------------- problem constants (match reference) ----------------
constexpr int HH = 192, WW = 192, DD = 512;
constexpr int KK = 5;
constexpr int NA = HH * WW * KK;          // 184320 anchors
constexpr float FIELD = 16.0f;
constexpr float TAU = 0.5f;
constexpr float NMS_IOU = 0.3f;
constexpr int MAX_BOXES = 256;
constexpr float NEGF = -1000000000.0f;    // NEG
constexpr int OUT_W = 3 * 3 * DD + 4 + 1; // 4613 per box

typedef float v2f __attribute__((ext_vector_type(2)));
typedef float v8f __attribute__((ext_vector_type(8)));

// =====================================================================
// Kernel 1: per-anchor gather + softmax + box decode + score masking
// =====================================================================
__global__ void decode_kernel(const float* __restrict__ obj,
                              const float* __restrict__ reg,
                              const float* __restrict__ anchors,
                              float* __restrict__ boxes,
                              float* __restrict__ smask,
                              float* __restrict__ sraw) {
  int a = blockIdx.x * blockDim.x + threadIdx.x;
  if (a >= NA) return;
  int k = a % KK;
  int j = (a / KK) % WW;
  int i = a / (KK * WW);

  const float* an = anchors + a * 4;
  float ax = an[0], ay = an[1], aw = an[2], ah = an[3];

  int cell = i * WW + j;
  float o0 = obj[cell * (2 * KK) + k * 2 + 0];
  float o1 = obj[cell * (2 * KK) + k * 2 + 1];
  float sc = 1.0f / (1.0f + expf(o0 - o1));   // softmax(...)[:,1]

  const float* rg = reg + cell * (4 * KK) + k * 4;
  float cx = ax + aw * 0.5f;
  float cy = ay + ah * 0.5f;
  float pcx = rg[0] * aw + cx;
  float pcy = rg[1] * ah + cy;
  float pw = aw * expf(rg[2]);
  float ph = ah * expf(rg[3]);

  boxes[a * 4 + 0] = pcx - pw * 0.5f;
  boxes[a * 4 + 1] = pcy - ph * 0.5f;
  boxes[a * 4 + 2] = pw;
  boxes[a * 4 + 3] = ph;
  sraw[a] = sc;
  smask[a] = (sc > TAU) ? sc : NEGF;
}

// =====================================================================
// Kernel 2: greedy NMS, single workgroup, 256 sequential rounds
// =====================================================================
__global__ void nms_kernel(const float* __restrict__ boxes,
                           float* __restrict__ s,
                           int* __restrict__ idxs,
                           int* __restrict__ valid) {
  __shared__ float rs[1024];
  __shared__ int   ri[1024];
  __shared__ float bb[5]; // x1,y1,x2,y2,area of the selected box
  const int tid = threadIdx.x;

  for (int it = 0; it < MAX_BOXES; ++it) {
    // ---- argmax over masked scores (first-occurrence tie-break) ----
    float bsv = -3.402823e38f;
    int   bii = 0x7fffffff;
    for (int a = tid; a < NA; a += 1024) {
      float v = s[a];
      if (v > bsv) { bsv = v; bii = a; }   // strict > keeps smallest index
    }
    rs[tid] = bsv; ri[tid] = bii;
    __syncthreads();
    for (int off = 512; off > 0; off >>= 1) {
      if (tid < off) {
        float v2 = rs[tid + off]; int i2 = ri[tid + off];
        if (v2 > rs[tid] || (v2 == rs[tid] && i2 < ri[tid])) { rs[tid] = v2; ri[tid] = i2; }
      }
      __syncthreads();
    }
    const int best = ri[0];
    if (tid == 0) {
      idxs[it]  = best;
      valid[it] = (rs[0] > NEGF * 0.5f) ? 1 : 0;
      float x1 = boxes[best * 4 + 0], y1 = boxes[best * 4 + 1];
      float w  = boxes[best * 4 + 2], h  = boxes[best * 4 + 3];
      bb[0] = x1; bb[1] = y1; bb[2] = x1 + w; bb[3] = y1 + h; bb[4] = w * h;
      s[best] = NEGF;
    }
    __syncthreads();
    // ---- IoU suppression against the selected box ----
    const float bx1 = bb[0], by1 = bb[1], bx2 = bb[2], by2 = bb[3], barea = bb[4];
    for (int a = tid; a < NA; a += 1024) {
      float x1 = boxes[a * 4 + 0], y1 = boxes[a * 4 + 1];
      float w  = boxes[a * 4 + 2], h  = boxes[a * 4 + 3];
      float ix1 = fmaxf(x1, bx1),     iy1 = fmaxf(y1, by1);
      float ix2 = fminf(x1 + w, bx2), iy2 = fminf(y1 + h, by2);
      float inter = fmaxf(ix2 - ix1, 0.0f) * fmaxf(iy2 - iy1, 0.0f);
      float iou = inter / (w * h + barea - inter + 1e-8f);
      if (iou > NMS_IOU) s[a] = NEGF;
    }
    __syncthreads();
  }
}

// =====================================================================
// Kernel 3: RoIAlign via V_WMMA_F32_16X16X4_F32 + 2x2 maxpool + assembly
//
// Per box, per pooled row-pair ty (3 of them):
//   D[12x16chan] = W[12x48] * G[48x16chan]
//   rows m = syl*6+sx  (syl in {0,1} -> sy = 2*ty+syl, sx in 0..5)
//   G rows k = syl*24 + yi*12 + xi  (yi: y0i/y1i neighbor, xi: x0i[0..5],x1i[0..5])
//   W[m,k] = wy(sy,yi) * wx(sx,xi): separable bilinear, branch-free selects.
// K=48 stepped as 12 WMMA(16x16x4) f32 accumulations (full f32 precision).
// =====================================================================

// Branch-free sparse bilinear weight. With the K loop unrolled, k is a
// compile-time function of (chunk, lane-half), so ks/yi/xi fold to selects.
__device__ __forceinline__ float wsel(int k, int syl, int sxc, bool validm,
                                      float axv, float ayv) {
  int ks  = k / 24;
  int kr  = k - ks * 24;
  int yi  = kr / 12;
  int xi  = kr - yi * 12;
  int xim = (xi < 6) ? xi : xi - 6;
  float wx = (xi < 6) ? (1.0f - axv) : axv;
  float wy = yi ? ayv : (1.0f - ayv);
  bool on = validm & (ks == syl) & (xim == sxc);
  return on ? wy * wx : 0.0f;
}

__global__ void roialign_kernel(const float* __restrict__ feat,
                                const float* __restrict__ boxes,
                                const float* __restrict__ sraw,
                                const int* __restrict__ idxs,
                                const int* __restrict__ valid,
                                float* __restrict__ out) {
  __shared__ float s_ax[6], s_ay[6];
  __shared__ int   s_x0[6], s_x1[6], s_y0[6], s_y1[6];
  __shared__ int   s_row[3][48];   // per-ty gathered-row base offsets
  __shared__ float s_box[4];
  __shared__ float s_vm, s_sc;

  const int b    = blockIdx.x;
  const int tid  = threadIdx.x;
  const int lane = tid & 31;
  const int wave = tid >> 5;           // 8 waves of 32 (wave32)

  if (tid == 0) {
    int idx = idxs[b];
    s_vm = valid[b] ? 1.0f : 0.0f;
    s_sc = sraw[idx];
    for (int q = 0; q < 4; ++q) s_box[q] = boxes[idx * 4 + q];
  }
  __syncthreads();
  if (tid < 6) {
    float t  = ((float)tid + 0.5f) / 6.0f;
    float fx = (s_box[0] + t * s_box[2]) / FIELD - 0.5f;
    float fy = (s_box[1] + t * s_box[3]) / FIELD - 0.5f;
    fx = fminf(fmaxf(fx, 0.0f), (float)(WW - 1));
    fy = fminf(fmaxf(fy, 0.0f), (float)(HH - 1));
    int x0 = (int)floorf(fx);
    int y0 = (int)floorf(fy);
    s_x0[tid] = x0;
    s_y0[tid] = y0;
    s_x1[tid] = (x0 + 1 < WW - 1) ? x0 + 1 : WW - 1;
    s_y1[tid] = (y0 + 1 < HH - 1) ? y0 + 1 : HH - 1;
    s_ax[tid] = fx - (float)x0;
    s_ay[tid] = fy - (float)y0;
  }
  __syncthreads();
  // precompute all 3*48 gathered-row base offsets (one thread per entry)
  if (tid < 144) {
    int ty  = tid / 48;
    int k   = tid - ty * 48;
    int syl = k / 24;
    int kr  = k - syl * 24;
    int yi  = kr / 12;
    int xi  = kr - yi * 12;
    int sy  = 2 * ty + syl;
    int y   = yi ? s_y1[sy] : s_y0[sy];
    int x   = (xi < 6) ? s_x0[xi] : s_x1[xi - 6];
    s_row[ty][k] = (y * WW + x) * DD;
  }
  __syncthreads();

  const float vm = s_vm;
  const int hi = (lane >> 4) & 1;      // lane half selects K rows {0,1} vs {2,3}
  const int m  = lane & 15;            // A: row per lane; B/D: column per lane
  const bool validm = (m < 12);
  const int  mc  = validm ? m : 0;
  const int  syl = mc / 6;             // 0 or 1: which sy of the pair
  const int  sxc = mc - syl * 6;       // 0..5
  const float axv = s_ax[sxc];         // hoisted: fixed per lane

  for (int ty = 0; ty < 3; ++ty) {
    const float ayv = s_ay[2 * ty + syl];          // hoisted per ty
    const int* __restrict__ rowt = s_row[ty];
    for (int tile = wave; tile < DD / 16; tile += 8) {   // uniform: 4 tiles/wave
      const int d0 = tile * 16;
      v8f acc = {0.f, 0.f, 0.f, 0.f, 0.f, 0.f, 0.f, 0.f};
      #pragma unroll
      for (int ch = 0; ch < 12; ++ch) {                  // K = 48 in steps of 4
        const int ka0 = ch * 4 + (hi ? 2 : 0);
        const int ka1 = ka0 + 1;
        v2f A, B;
        A.x = wsel(ka0, syl, sxc, validm, axv, ayv);
        A.y = wsel(ka1, syl, sxc, validm, axv, ayv);
        B.x = feat[rowt[ka0] + d0 + m];
        B.y = feat[rowt[ka1] + d0 + m];
        acc = __builtin_amdgcn_wmma_f32_16x16x4_f32(
            /*neg_a=*/false, A, /*neg_b=*/false, B,
            /*c_mod=*/(short)0, acc, /*reuse_a=*/false, /*reuse_b=*/false);
      }
      // ---- 2x2 maxpool over D rows {2tx,2tx+1,6+2tx,7+2tx} ----
      // low lanes hold rows 0..7 in acc[0..7]; high lanes rows 8..15
      float m01 = fmaxf(acc[0], acc[1]);
      float m23 = fmaxf(acc[2], acc[3]);
      float m45 = fmaxf(acc[4], acc[5]);
      float m67 = fmaxf(acc[6], acc[7]);
      float o1 = __shfl_xor(m01, 16, 32);  // rows 8,9 from high half
      float o2 = __shfl_xor(m23, 16, 32);  // rows 10,11 from high half
      if (lane < 16) {
        float t0 = fmaxf(m01, m67);        // rows 0,1,6,7   -> tx=0
        float t1 = fmaxf(m23, o1);         // rows 2,3,8,9   -> tx=1
        float t2 = fmaxf(m45, o2);         // rows 4,5,10,11 -> tx=2
        int base = b * OUT_W + (ty * 3) * DD + d0 + lane;
        out[base + 0 * DD] = vm * t0;
        out[base + 1 * DD] = vm * t1;
        out[base + 2 * DD] = vm * t2;
      }
    }
  }

  if (tid < 4)  out[b * OUT_W + 3 * 3 * DD + tid] = vm * s_box[tid];
  if (tid == 4) out[b * OUT_W + 3 * 3 * DD + 4]   = vm * s_sc;
}

// =====================================================================
// launch: d_in = {features, output_obj, output_reg, anchors, anchor_px}
// ws layout (floats): boxes[4N] | smask[N] | sraw[N] | idxs[256] | valid[256]
// (needs ~4.43 MB of scratch; everything is rewritten each call)
// =====================================================================
extern "C" void kernel_launch(void* const* d_in, const int* in_sizes, int n_in,
                              void* d_out, int out_size, void* d_ws, size_t ws_size,
                              hipStream_t stream) {
  const float* features = (const float*)d_in[0];
  const float* obj      = (const float*)d_in[1];
  const float* reg      = (const float*)d_in[2];
  const float* anchors  = (const float*)d_in[3];
  (void)d_in[4]; (void)in_sizes; (void)n_in; (void)out_size; (void)ws_size;

  float* ws    = (float*)d_ws;
  float* boxes = ws;                 // 4*NA
  float* smask = ws + 4 * NA;        // NA
  float* sraw  = ws + 5 * NA;        // NA
  int*   idxs  = (int*)(ws + 6 * NA);          // 256
  int*   valid = (int*)(ws + 6 * NA + 256);    // 256
  float* out   = (float*)d_out;

  decode_kernel<<<NA / 256, 256, 0, stream>>>(obj, reg, anchors, boxes, smask, sraw);
  nms_kernel<<<1, 1024, 0, stream>>>(boxes, smask, idxs, valid);
  roialign_kernel<<<MAX_BOXES, 256, 0, stream>>>(features, boxes, sraw, idxs, valid, out);
}